// SAGPooling_20126216750003
// MI455X (gfx1250) — compile-verified
//
#include <hip/hip_runtime.h>
#include <hip/hip_bf16.h>

typedef __attribute__((ext_vector_type(2))) float v2f;
typedef __attribute__((ext_vector_type(8))) float v8f;

#define THREADS 256
#define N_GRAPHS 100
#define SORTN 2048
#define EDGES_PER_THREAD 8
#define EDGES_PER_BLOCK (THREADS * EDGES_PER_THREAD)

// ---------------------------------------------------------------------------
// 0) Per-call re-init: agg = 0, nmask = -1 (harness does not re-poison ws)
// ---------------------------------------------------------------------------
__global__ void init_kernel(float* __restrict__ agg, int* __restrict__ nmask, int N) {
    int i = blockIdx.x * blockDim.x + threadIdx.x;
    if (i < N) { agg[i] = 0.0f; nmask[i] = -1; }
}

// ---------------------------------------------------------------------------
// 1) Dual mat-vec via WMMA: s[i] = x[i]·w_rel, rdot[i] = x[i]·w_root.
//    One wave handles 32 nodes with two accumulators sharing one B tile.
//    B(4x16) has w_rel in col 0, w_root in col 1, zeros elsewhere; inactive
//    lanes read a zero-filled LDS section so the DS load is branchless
//    (no exec-mask save/restore in the WMMA hot loop).
// ---------------------------------------------------------------------------
__global__ void score_wmma_kernel(const float* __restrict__ x,
                                  const float* __restrict__ w_rel,
                                  const float* __restrict__ w_root,
                                  float* __restrict__ s,
                                  float* __restrict__ rdot,
                                  int N) {
    __shared__ float wlds[768];     // [0,256): w_rel  [256,512): w_root  [512,768): zeros
    for (int i = threadIdx.x; i < 256; i += blockDim.x) {
        wlds[i]       = w_rel[i];
        wlds[256 + i] = w_root[i];
        wlds[512 + i] = 0.0f;
    }
    __syncthreads();

    const int wave = blockIdx.x * (blockDim.x >> 5) + (threadIdx.x >> 5);
    const int base = wave * 32;
    if (base >= N) return;                      // wave-uniform guard (N % 32 == 0)

    const int lane = threadIdx.x & 31;
    const int mrow = lane & 15;                 // A: matrix row;  B: column index N
    const int grp  = lane >> 4;                 // K sub-group: {0,1} vs {2,3}

    // A-matrix streams: lane holds x[row][4t + 2*grp + {0,1}] for two 16-row tiles
    const float* px0 = x + (size_t)(base + mrow) * 256 + (size_t)(grp * 2);
    const float* px1 = px0 + (size_t)16 * 256;
    // Branchless B feed: columns >= 2 read architectural zeros from LDS
    const int sel = (mrow < 2) ? mrow : 2;
    const float* pw = &wlds[sel * 256 + grp * 2];

    v8f c0 = {};
    v8f c1 = {};
    for (int t = 0; t < 64; ++t) {
        if ((t & 3) == 0) {
            __builtin_prefetch(px0 + 16, 0, 3);     // global_prefetch_b8
            __builtin_prefetch(px1 + 16, 0, 3);
        }
        v2f a0; a0.x = px0[0]; a0.y = px0[1]; px0 += 4;
        v2f a1; a1.x = px1[0]; a1.y = px1[1]; px1 += 4;
        v2f b;  b.x  = pw[0];  b.y  = pw[1];  pw  += 4;
        // (neg_a, A, neg_b, B, c_mod, C, reuse_a, reuse_b)
        c0 = __builtin_amdgcn_wmma_f32_16x16x4_f32(false, a0, false, b,
                                                   (short)0, c0, false, false);
        c1 = __builtin_amdgcn_wmma_f32_16x16x4_f32(false, a1, false, b,
                                                   (short)0, c1, false, false);
    }

    // D layout: VGPR v, lane l -> M = v + 8*(l>>4), N = l&15.
    // Column 0 (w_rel dots) lives in lanes 0/16; column 1 in lanes 1/17.
    if (mrow == 0) {
        for (int v = 0; v < 8; ++v) {
            s[base +      grp * 8 + v] = c0[v];
            s[base + 16 + grp * 8 + v] = c1[v];
        }
    } else if (mrow == 1) {
        for (int v = 0; v < 8; ++v) {
            rdot[base +      grp * 8 + v] = c0[v];
            rdot[base + 16 + grp * 8 + v] = c1[v];
        }
    }
}

// ---------------------------------------------------------------------------
// 2) Edge aggregation: agg[col[e]] += s[row[e]]   (global_atomic_add_f32)
// ---------------------------------------------------------------------------
__global__ void aggregate_kernel(const long long* __restrict__ row,
                                 const long long* __restrict__ col,
                                 const float* __restrict__ s,
                                 float* __restrict__ agg,
                                 int E) {
    int stride = gridDim.x * blockDim.x;
    for (int e = blockIdx.x * blockDim.x + threadIdx.x; e < E; e += stride) {
        atomicAdd(&agg[col[e]], s[row[e]]);
    }
}

// ---------------------------------------------------------------------------
// 3) score = agg + rdot + b; per-graph descending bitonic sort (stable ties),
//    emit perm (top-k global node ids). One block per graph, 2048-key LDS sort.
// ---------------------------------------------------------------------------
__global__ void topk_kernel(const float* __restrict__ agg,
                            const float* __restrict__ rdot,
                            const float* __restrict__ b_rel,
                            float* __restrict__ score,
                            int* __restrict__ perm,
                            int PG, int K) {
    __shared__ unsigned long long keys[SORTN];
    const int g   = blockIdx.x;
    const int tid = threadIdx.x;
    const float b = b_rel[0];

    for (int i = tid; i < SORTN; i += THREADS) {
        unsigned long long key = 0ull;          // pad: smallest key, never in top-k
        if (i < PG) {
            int node = g * PG + i;
            float sc = agg[node] + rdot[node] + b;
            score[node] = sc;
            unsigned u = __float_as_uint(sc);
            u = (u & 0x80000000u) ? ~u : (u | 0x80000000u);   // order-preserving map
            // descending score, then ascending index (stable-argsort semantics)
            key = ((unsigned long long)u << 32) | (unsigned)(0xFFFFFFFFu - (unsigned)i);
        }
        keys[i] = key;
    }
    __syncthreads();

    for (int k = 2; k <= SORTN; k <<= 1) {
        for (int j = k >> 1; j > 0; j >>= 1) {
            for (int i = tid; i < SORTN; i += THREADS) {
                int ixj = i ^ j;
                if (ixj > i) {
                    bool desc = ((i & k) == 0);
                    unsigned long long a0 = keys[i], a1 = keys[ixj];
                    if (desc ? (a0 < a1) : (a0 > a1)) { keys[i] = a1; keys[ixj] = a0; }
                }
            }
            __syncthreads();
        }
    }

    for (int i = tid; i < K; i += THREADS) {
        unsigned long long e = keys[i];
        int li = (int)(0xFFFFFFFFu - (unsigned)(e & 0xFFFFFFFFull));
        perm[g * K + i] = g * PG + li;
    }
}

// ---------------------------------------------------------------------------
// 4) nmask[perm[j]] = j
// ---------------------------------------------------------------------------
__global__ void scatter_kernel(const int* __restrict__ perm,
                               int* __restrict__ nmask, int M) {
    int j = blockIdx.x * blockDim.x + threadIdx.x;
    if (j < M) nmask[perm[j]] = j;
}

// ---------------------------------------------------------------------------
// 5) Edge compaction: per-block keep-counts
// ---------------------------------------------------------------------------
__global__ void edge_count_kernel(const long long* __restrict__ row,
                                  const long long* __restrict__ col,
                                  const int* __restrict__ nmask,
                                  int* __restrict__ bcount, int E) {
    __shared__ int cnts[THREADS];
    const int t = threadIdx.x;
    long long base = (long long)blockIdx.x * EDGES_PER_BLOCK + (long long)t * EDGES_PER_THREAD;
    int c = 0;
    for (int k = 0; k < EDGES_PER_THREAD; ++k) {
        long long e = base + k;
        if (e < E) {
            int r2 = nmask[row[e]];
            int c2 = nmask[col[e]];
            if (r2 >= 0 && c2 >= 0) c++;
        }
    }
    cnts[t] = c;
    __syncthreads();
    for (int off = THREADS >> 1; off > 0; off >>= 1) {
        if (t < off) cnts[t] += cnts[t + off];
        __syncthreads();
    }
    if (t == 0) bcount[blockIdx.x] = cnts[0];
}

// 6) Exclusive scan of block counts (small: ~2K blocks), total kept-edge count
__global__ void scan_kernel(const int* __restrict__ bcount,
                            int* __restrict__ boffset,
                            int* __restrict__ total, int nblk) {
    if (blockIdx.x == 0 && threadIdx.x == 0) {
        int run = 0;
        for (int i = 0; i < nblk; ++i) { boffset[i] = run; run += bcount[i]; }
        total[0] = run;
    }
}

// 7) Order-preserving compacted write of remapped edges into d_out
__global__ void edge_write_kernel(const long long* __restrict__ row,
                                  const long long* __restrict__ col,
                                  const int* __restrict__ nmask,
                                  const int* __restrict__ boffset,
                                  const int* __restrict__ total,
                                  char* __restrict__ outBase,
                                  long long xbytes, int E) {
    __shared__ int cnts[THREADS];
    const int t = threadIdx.x;
    long long base = (long long)blockIdx.x * EDGES_PER_BLOCK + (long long)t * EDGES_PER_THREAD;
    int c = 0;
    for (int k = 0; k < EDGES_PER_THREAD; ++k) {
        long long e = base + k;
        if (e < E) {
            int r2 = nmask[row[e]];
            int c2 = nmask[col[e]];
            if (r2 >= 0 && c2 >= 0) c++;
        }
    }
    cnts[t] = c;
    __syncthreads();
    for (int off = 1; off < THREADS; off <<= 1) {      // Hillis-Steele inclusive scan
        int v = (t >= off) ? cnts[t - off] : 0;
        __syncthreads();
        cnts[t] += v;
        __syncthreads();
    }
    int pos = boffset[blockIdx.x] + cnts[t] - c;        // exclusive start for thread
    long long Ep = total[0];
    long long* erow = (long long*)(outBase + xbytes);
    long long* ecol = erow + Ep;
    for (int k = 0; k < EDGES_PER_THREAD; ++k) {
        long long e = base + k;
        if (e < E) {
            int r2 = nmask[row[e]];
            int c2 = nmask[col[e]];
            if (r2 >= 0 && c2 >= 0) { erow[pos] = (long long)r2; ecol[pos] = (long long)c2; pos++; }
        }
    }
}

// ---------------------------------------------------------------------------
// 8) Gated gather: x_out[j] = x[perm[j]] * tanh(score[perm[j]]);
//    also batch_out and perm (int64) placed after the variable edge section.
// ---------------------------------------------------------------------------
__global__ void gather_kernel(const float* __restrict__ x,
                              const float* __restrict__ score,
                              const int* __restrict__ perm,
                              const long long* __restrict__ batch,
                              char* __restrict__ outBase,
                              long long xbytes,
                              const int* __restrict__ total,
                              int M) {
    const int wave = blockIdx.x * (blockDim.x >> 5) + (threadIdx.x >> 5);
    if (wave >= M) return;
    const int lane = threadIdx.x & 31;
    const int node = perm[wave];
    const float gate = tanhf(score[node]);

    const float4* src = (const float4*)(x + (size_t)node * 256);
    float4* dst = (float4*)((float*)outBase + (size_t)wave * 256);
    for (int i = lane; i < 64; i += 32) {
        float4 v = src[i];
        v.x *= gate; v.y *= gate; v.z *= gate; v.w *= gate;
        dst[i] = v;
    }
    if (lane == 0) {
        long long Ep = total[0];
        long long* batch_out = (long long*)(outBase + xbytes + 16ll * Ep);
        long long* perm64   = batch_out + M;
        batch_out[wave] = batch[node];
        perm64[wave]    = (long long)node;
    }
}

// ---------------------------------------------------------------------------
// Host-side launcher
// ---------------------------------------------------------------------------
extern "C" void kernel_launch(void* const* d_in, const int* in_sizes, int n_in,
                              void* d_out, int out_size, void* d_ws, size_t ws_size,
                              hipStream_t stream) {
    const float*     x      = (const float*)d_in[0];
    const long long* edge   = (const long long*)d_in[1];   // int64 in reference
    const long long* batch  = (const long long*)d_in[2];
    const float*     w_root = (const float*)d_in[3];
    const float*     w_rel  = (const float*)d_in[4];
    const float*     b_rel  = (const float*)d_in[5];

    const int N  = in_sizes[2];          // 200000
    const int E  = in_sizes[1] / 2;      // 4000000
    const int G  = N_GRAPHS;             // 100 (reference constant)
    const int PG = N / G;                // 2000 nodes per graph
    const int K  = (PG + 1) / 2;         // ceil(RATIO * PG) = 1000
    const int M  = G * K;                // 100000 kept nodes

    const long long* row = edge;
    const long long* col = edge + E;

    // workspace carve-up (~4.5 MB)
    float* s      = (float*)d_ws;
    float* rdot   = s + N;
    float* agg    = rdot + N;
    float* score  = agg + N;
    int*   nmask  = (int*)(score + N);
    int*   perm   = nmask + N;
    int*   bcount = perm + M;
    const int NBLK = (E + EDGES_PER_BLOCK - 1) / EDGES_PER_BLOCK;
    int*   boffset = bcount + NBLK;
    int*   total   = boffset + NBLK;

    const long long xbytes = (long long)M * 256 * (long long)sizeof(float);

    // 0) re-init agg / nmask
    init_kernel<<<(N + THREADS - 1) / THREADS, THREADS, 0, stream>>>(agg, nmask, N);

    // 1) WMMA dual mat-vec (32 nodes per wave, 8 waves per block)
    {
        int nWaves = (N + 31) / 32;
        int blocks = (nWaves + 7) / 8;
        score_wmma_kernel<<<blocks, THREADS, 0, stream>>>(x, w_rel, w_root, s, rdot, N);
    }

    // 2) edge aggregation
    aggregate_kernel<<<4096, THREADS, 0, stream>>>(row, col, s, agg, E);

    // 3) per-graph top-k (score fused in)
    topk_kernel<<<G, THREADS, 0, stream>>>(agg, rdot, b_rel, score, perm, PG, K);

    // 4) node-id remap table
    scatter_kernel<<<(M + THREADS - 1) / THREADS, THREADS, 0, stream>>>(perm, nmask, M);

    // 5-7) order-preserving edge compaction
    edge_count_kernel<<<NBLK, THREADS, 0, stream>>>(row, col, nmask, bcount, E);
    scan_kernel<<<1, THREADS, 0, stream>>>(bcount, boffset, total, NBLK);
    edge_write_kernel<<<NBLK, THREADS, 0, stream>>>(row, col, nmask, boffset, total,
                                                    (char*)d_out, xbytes, E);

    // 8) gated gather + batch_out + perm64
    gather_kernel<<<(M + 7) / 8, THREADS, 0, stream>>>(x, score, perm, batch,
                                                       (char*)d_out, xbytes, total, M);
}